// Module_GAT_DET_49254684950641
// MI455X (gfx1250) — compile-verified
//
#include <hip/hip_runtime.h>
#include <hip/hip_bf16.h>
#include <math.h>

// ---------------------------------------------------------------------------
// GAT layer for MI455X (gfx1250), fp32 via V_WMMA_F32_16X16X4_F32.
// B=2, N=1024, D=128, DA=32, NH=4, AH=32, DEC_H=256, DEC_O=128.
// Attention stage double-buffers its tiles with GLOBAL_LOAD_ASYNC_TO_LDS_B128
// (ASYNCcnt) so the DMA of tile j+1 overlaps score-VALU + WMMA of tile j.
// ---------------------------------------------------------------------------

typedef __attribute__((ext_vector_type(2))) float v2f;
typedef __attribute__((ext_vector_type(4))) float f4;
typedef __attribute__((ext_vector_type(8))) float v8f;

#define BN    2048      // B*N
#define NSEQ  1024
#define DD    128
#define DAH   32        // DA
#define NHEAD 4
#define AHID  32        // AH
#define DECH  256
#define DECO  128
#define CDIM  640       // (NH+1)*D
#define MASKC 10000.0f

// ---- WMMA helpers ----------------------------------------------------------

__device__ __forceinline__ v8f wmma4(v2f a, v2f b, v8f c) {
  // D = A(16x4 f32) * B(4x16 f32) + C(16x16 f32)
  return __builtin_amdgcn_wmma_f32_16x16x4_f32(
      /*neg_a=*/false, a, /*neg_b=*/false, b,
      /*c_mod=*/(short)0, c, /*reuse_a=*/false, /*reuse_b=*/false);
}

// A fragment: row-major A already offset to (m0,k0); lda in floats.
// lane m = lane&15 ; half h = lane>>4 picks K = {2h, 2h+1}.
__device__ __forceinline__ v2f frag_a(const float* A, int lda, int lane) {
  int m = lane & 15, h = lane >> 4;
  v2f a;
  a.x = A[m * lda + 2 * h];
  a.y = A[m * lda + 2 * h + 1];
  return a;
}

// B fragment: row-major B already offset to (k0,n0); ldb in floats.
__device__ __forceinline__ v2f frag_b(const float* Bm, int ldb, int lane) {
  int n = lane & 15, h = lane >> 4;
  v2f b;
  b.x = Bm[(2 * h) * ldb + n];
  b.y = Bm[(2 * h + 1) * ldb + n];
  return b;
}

// ---- CDNA5 async global->LDS DMA (ASYNCcnt) --------------------------------

// One wave-wide async b128: 32 lanes x 16B = 512B LDS fill, no VGPR staging.
__device__ __forceinline__ void async_copy_b128(const f4* gsrc, unsigned lds_off) {
  asm volatile("global_load_async_to_lds_b128 %0, %1, off"
               :: "v"(lds_off), "v"(gsrc)
               : "memory");
}
__device__ __forceinline__ void wait_asynccnt0() {
  asm volatile("s_wait_asynccnt 0x0" ::: "memory");
}
__device__ __forceinline__ void wait_dscnt0() {
  asm volatile("s_wait_dscnt 0x0" ::: "memory");
}
__device__ __forceinline__ unsigned lds_addr_of(const void* p) {
  // Shared-memory flat address: aperture in [63:32], LDS offset in [31:0].
  return (unsigned)(uintptr_t)p;
}

// ---- Kernel 1: k/q/v = x @ W + b ; v additionally copied into c[:,0:128] ---

__global__ __launch_bounds__(32)
void qkv_kernel(const float* __restrict__ x,
                const float* __restrict__ Wk, const float* __restrict__ bk,
                const float* __restrict__ Wq, const float* __restrict__ bq,
                const float* __restrict__ Wv, const float* __restrict__ bv,
                float* __restrict__ kbuf, float* __restrict__ qbuf,
                float* __restrict__ vbuf, float* __restrict__ cbuf) {
  const int lane  = threadIdx.x;
  const int mt    = blockIdx.x;   // 0..127  (BN/16)
  const int nt    = blockIdx.y;   // 0..7    (D/16)
  const int which = blockIdx.z;   // 0:k 1:q 2:v

  const float* W    = (which == 0) ? Wk : (which == 1) ? Wq : Wv;
  const float* bias = (which == 0) ? bk : (which == 1) ? bq : bv;
  float*       out  = (which == 0) ? kbuf : (which == 1) ? qbuf : vbuf;

  const int n = lane & 15, h = lane >> 4;
  const float bini = bias[nt * 16 + n];
  v8f acc;
#pragma unroll
  for (int r = 0; r < 8; ++r) acc[r] = bini;

  const float* Abase = x + mt * 16 * DD;
  const float* Bbase = W + nt * 16;
#pragma unroll 4
  for (int k0 = 0; k0 < DD; k0 += 4) {
    v2f a = frag_a(Abase + k0, DD, lane);
    v2f b = frag_b(Bbase + k0 * DD, DD, lane);
    acc = wmma4(a, b, acc);
  }

  float* o = out + (mt * 16) * DD + nt * 16;
#pragma unroll
  for (int r = 0; r < 8; ++r) o[(r + 8 * h) * DD + n] = acc[r];
  if (which == 2) {
    float* o2 = cbuf + (mt * 16) * CDIM + nt * 16;  // c[:, 0:128] = v
#pragma unroll
    for (int r = 0; r < 8; ++r) o2[(r + 8 * h) * CDIM + n] = acc[r];
  }
}

// ---- Kernel 2: hk = k_head @ Wa1[:32] ; hq = q_head @ Wa1[32:] + ba1 -------

__global__ __launch_bounds__(32)
void attnproj_kernel(const float* __restrict__ kbuf, const float* __restrict__ qbuf,
                     const float* __restrict__ Wa1, const float* __restrict__ ba1,
                     float* __restrict__ hkbuf, float* __restrict__ hqbuf) {
  const int lane = threadIdx.x;
  const int mt   = blockIdx.x;       // 0..127
  const int nt   = blockIdx.y;       // 0..1  (AH/16)
  const int z    = blockIdx.z;       // 0..7 : head*2 + isq
  const int head = z >> 1, isq = z & 1;

  const float* src = isq ? qbuf : kbuf;
  float*       dst = isq ? hqbuf : hkbuf;

  const int n = lane & 15, h = lane >> 4;
  const float bini = isq ? ba1[nt * 16 + n] : 0.0f;  // fold ba1 into hq
  v8f acc;
#pragma unroll
  for (int r = 0; r < 8; ++r) acc[r] = bini;

  const float* Abase = src + mt * 16 * DD + head * DAH;
  const float* Bbase = Wa1 + (isq ? DAH : 0) * AHID + nt * 16;
#pragma unroll
  for (int k0 = 0; k0 < DAH; k0 += 4) {
    v2f a = frag_a(Abase + k0, DD, lane);
    v2f b = frag_b(Bbase + k0 * AHID, AHID, lane);
    acc = wmma4(a, b, acc);
  }

  float* o = dst + (head * BN + mt * 16) * AHID + nt * 16;
#pragma unroll
  for (int r = 0; r < 8; ++r) o[(r + 8 * h) * AHID + n] = acc[r];
}

// ---- Kernel 3: scores -> sigmoid -> w @ v (flash-style, async double-buf) --

__global__ __launch_bounds__(32)
void attn_kernel(const float* __restrict__ hkbuf, const float* __restrict__ hqbuf,
                 const float* __restrict__ vbuf, const float* __restrict__ Wa2,
                 const float* __restrict__ ba2, float* __restrict__ cbuf) {
  __shared__ float s_hk[16 * AHID];      // 2 KB   (persistent i-tile)
  __shared__ float s_hq[2][16 * AHID];   // 4 KB   (double-buffered j-tiles)
  __shared__ float s_w2[AHID];
  __shared__ float s_w[16 * 16];         // 1 KB   (w tile, A-matrix of w@v)
  __shared__ float s_v[2][16 * DD];      // 16 KB  (double-buffered v j-tiles)

  const int lane = threadIdx.x;
  const int it   = blockIdx.x;   // 0..63  (i tile within sequence)
  const int head = blockIdx.y;   // 0..3
  const int b    = blockIdx.z;   // 0..1

  // preload hk i-tile + Wa2 (persistent for this wave)
  {
    const float* src = hkbuf + (head * BN + b * NSEQ + it * 16) * AHID;
    for (int idx = lane; idx < 16 * AHID; idx += 32) s_hk[idx] = src[idx];
    s_w2[lane] = Wa2[lane];   // AH == 32 == wave width
  }
  const float ba2v = ba2[0];

  const float* vrow  = vbuf + (b * NSEQ) * DD;
  const float* hqrow = hqbuf + (head * BN + b * NSEQ) * AHID;

  // Issue the async DMA for tile `jc` into buffer `bi`:
  //  v tile: 16x128 f32 = 8 KB  -> 16 wave-wide b128 ops
  //  hq tile: 16x32 f32 = 2 KB  ->  4 wave-wide b128 ops  (20 in flight < 63)
  auto issue_tile = [&](int jc, int bi) {
    const f4* vg = (const f4*)(vrow + jc * 16 * DD);
    unsigned  vl = lds_addr_of(&s_v[bi][0]);
#pragma unroll
    for (int t = 0; t < 16; ++t) {
      int idx = t * 32 + lane;
      async_copy_b128(vg + idx, vl + idx * 16);
    }
    const f4* hg = (const f4*)(hqrow + jc * 16 * AHID);
    unsigned  hl = lds_addr_of(&s_hq[bi][0]);
#pragma unroll
    for (int t = 0; t < 4; ++t) {
      int idx = t * 32 + lane;
      async_copy_b128(hg + idx, hl + idx * 16);
    }
  };

  v8f acc[8];
#pragma unroll
  for (int dt = 0; dt < 8; ++dt)
#pragma unroll
    for (int r = 0; r < 8; ++r) acc[dt][r] = 0.0f;

  const int il   = lane & 15;   // score row this lane owns
  const int hgrp = lane >> 4;   // score column group (8 cols)
  const int gi   = it * 16 + il;

  issue_tile(0, 0);
  wait_asynccnt0();

  for (int jc = 0; jc < NSEQ / 16; ++jc) {
    const int cur = jc & 1;
    if (jc + 1 < NSEQ / 16) {
      wait_dscnt0();               // DS reads of buf cur^1 from iter jc-1 done
      issue_tile(jc + 1, cur ^ 1); // overlap DMA with score VALU + WMMA below
    }
    const float* hq_t = s_hq[cur];
    const float* v_t  = s_v[cur];

    // VALU stage: s[i][j] = sum_a relu(hk+hq)*Wa2  (ba1 folded into hq)
    float sc[8];
#pragma unroll
    for (int t = 0; t < 8; ++t) sc[t] = 0.0f;
#pragma unroll 8
    for (int a = 0; a < AHID; ++a) {
      const float hk_ = s_hk[il * AHID + a];
      const float w2  = s_w2[a];
#pragma unroll
      for (int t = 0; t < 8; ++t) {
        const float hq_ = hq_t[(hgrp * 8 + t) * AHID + a];
        sc[t] = fmaf(fmaxf(hk_ + hq_, 0.0f), w2, sc[t]);
      }
    }
#pragma unroll
    for (int t = 0; t < 8; ++t) {
      const int j  = hgrp * 8 + t;
      const int gj = jc * 16 + j;
      const float s = sc[t] + ba2v - ((gi == gj) ? MASKC : 0.0f);
      s_w[il * 16 + j] = 1.0f / (1.0f + __expf(-s));  // sigmoid(s - mask)
    }
    __syncthreads();  // single-wave WG: NOP; LDS ops are in-order per wave

    // WMMA stage: acc[dt] += w(16x16) @ v(16x16dt), K split into 4x k=4
#pragma unroll
    for (int k0 = 0; k0 < 16; k0 += 4) {
      v2f a = frag_a(s_w + k0, 16, lane);
#pragma unroll
      for (int dt = 0; dt < 8; ++dt) {
        v2f bb = frag_b(v_t + k0 * DD + dt * 16, DD, lane);
        acc[dt] = wmma4(a, bb, acc[dt]);
      }
    }

    if (jc + 1 < NSEQ / 16)
      wait_asynccnt0();            // next buffer resident before next iter
  }

  // write head output into c[:, 128*(1+head) : 128*(2+head)]
  const int n = lane & 15;
  float* o = cbuf + (b * NSEQ + it * 16) * CDIM + (1 + head) * DD;
#pragma unroll
  for (int dt = 0; dt < 8; ++dt)
#pragma unroll
    for (int r = 0; r < 8; ++r)
      o[(r + 8 * hgrp) * CDIM + dt * 16 + n] = acc[dt][r];
}

// ---- Kernel 4: h1 = relu(c @ Wd1 + bd1) ------------------------------------

__global__ __launch_bounds__(32)
void dec1_kernel(const float* __restrict__ cbuf, const float* __restrict__ Wd1,
                 const float* __restrict__ bd1, float* __restrict__ h1) {
  const int lane = threadIdx.x;
  const int mt = blockIdx.x;   // 0..127
  const int nt = blockIdx.y;   // 0..15 (DEC_H/16)
  const int n = lane & 15, h = lane >> 4;
  const float bini = bd1[nt * 16 + n];
  v8f acc;
#pragma unroll
  for (int r = 0; r < 8; ++r) acc[r] = bini;

  const float* Abase = cbuf + mt * 16 * CDIM;
  const float* Bbase = Wd1 + nt * 16;
#pragma unroll 4
  for (int k0 = 0; k0 < CDIM; k0 += 4) {
    v2f a = frag_a(Abase + k0, CDIM, lane);
    v2f b = frag_b(Bbase + k0 * DECH, DECH, lane);
    acc = wmma4(a, b, acc);
  }

  float* o = h1 + mt * 16 * DECH + nt * 16;
#pragma unroll
  for (int r = 0; r < 8; ++r)
    o[(r + 8 * h) * DECH + n] = fmaxf(acc[r], 0.0f);
}

// ---- Kernel 5: d = h1 @ Wd2 + bd2 ------------------------------------------

__global__ __launch_bounds__(32)
void dec2_kernel(const float* __restrict__ h1, const float* __restrict__ Wd2,
                 const float* __restrict__ bd2, float* __restrict__ dout) {
  const int lane = threadIdx.x;
  const int mt = blockIdx.x;   // 0..127
  const int nt = blockIdx.y;   // 0..7 (DEC_O/16)
  const int n = lane & 15, h = lane >> 4;
  const float bini = bd2[nt * 16 + n];
  v8f acc;
#pragma unroll
  for (int r = 0; r < 8; ++r) acc[r] = bini;

  const float* Abase = h1 + mt * 16 * DECH;
  const float* Bbase = Wd2 + nt * 16;
#pragma unroll 4
  for (int k0 = 0; k0 < DECH; k0 += 4) {
    v2f a = frag_a(Abase + k0, DECH, lane);
    v2f b = frag_b(Bbase + k0 * DECO, DECO, lane);
    acc = wmma4(a, b, acc);
  }

  float* o = dout + mt * 16 * DECO + nt * 16;
#pragma unroll
  for (int r = 0; r < 8; ++r) o[(r + 8 * h) * DECO + n] = acc[r];
}

// ---- Host launcher ---------------------------------------------------------

extern "C" void kernel_launch(void* const* d_in, const int* in_sizes, int n_in,
                              void* d_out, int out_size, void* d_ws, size_t ws_size,
                              hipStream_t stream) {
  const float* x   = (const float*)d_in[0];
  const float* Wk  = (const float*)d_in[1];
  const float* bk  = (const float*)d_in[2];
  const float* Wq  = (const float*)d_in[3];
  const float* bq  = (const float*)d_in[4];
  const float* Wv  = (const float*)d_in[5];
  const float* bv  = (const float*)d_in[6];
  const float* Wa1 = (const float*)d_in[7];
  const float* ba1 = (const float*)d_in[8];
  const float* Wa2 = (const float*)d_in[9];
  const float* ba2 = (const float*)d_in[10];
  const float* Wd1 = (const float*)d_in[11];
  const float* bd1 = (const float*)d_in[12];
  const float* Wd2 = (const float*)d_in[13];
  const float* bd2 = (const float*)d_in[14];
  float* out = (float*)d_out;

  // Workspace layout (floats): 12 MiB total, L2-resident on MI455X (192 MB).
  float* ws   = (float*)d_ws;
  float* kbuf = ws;                         // 2048*128
  float* qbuf = kbuf + BN * DD;             // 2048*128
  float* vbuf = qbuf + BN * DD;             // 2048*128
  float* hkb  = vbuf + BN * DD;             // 4*2048*32
  float* hqb  = hkb + NHEAD * BN * AHID;    // 4*2048*32
  float* cbuf = hqb + NHEAD * BN * AHID;    // 2048*640
  float* h1   = cbuf + BN * CDIM;           // 2048*256

  // 1) k,q,v projections (+ v copied into c[:,0:128])
  qkv_kernel<<<dim3(BN / 16, DD / 16, 3), 32, 0, stream>>>(
      x, Wk, bk, Wq, bq, Wv, bv, kbuf, qbuf, vbuf, cbuf);

  // 2) per-head hk / hq projections (ba1 folded into hq)
  attnproj_kernel<<<dim3(BN / 16, AHID / 16, NHEAD * 2), 32, 0, stream>>>(
      kbuf, qbuf, Wa1, ba1, hkb, hqb);

  // 3) attention scores + sigmoid + w @ v  (the FLOP-dominant stage)
  attn_kernel<<<dim3(NSEQ / 16, NHEAD, 2), 32, 0, stream>>>(
      hkb, hqb, vbuf, Wa2, ba2, cbuf);

  // 4) decoder layer 1 with ReLU
  dec1_kernel<<<dim3(BN / 16, DECH / 16), 32, 0, stream>>>(cbuf, Wd1, bd1, h1);

  // 5) decoder layer 2 -> output
  dec2_kernel<<<dim3(BN / 16, DECO / 16), 32, 0, stream>>>(h1, Wd2, bd2, out);
}